// MoPKL_73950746903020
// MI455X (gfx1250) — compile-verified
//
#include <hip/hip_runtime.h>
#include <math.h>

#define B_  4
#define F_  5
#define N_  2048
#define D_  256
#define BF_ (B_ * F_)

typedef __attribute__((ext_vector_type(2))) float        v2f;
typedef __attribute__((ext_vector_type(8))) float        v8f;
typedef __attribute__((ext_vector_type(4))) unsigned int v4u;
typedef __attribute__((ext_vector_type(4))) int          v4i;
typedef __attribute__((ext_vector_type(8))) int          v8i;

__device__ __forceinline__ float finf() { return __int_as_float(0x7f800000); }

// float atomic-min via signed/unsigned integer ordering trick (valid with +INF init)
__device__ __forceinline__ void atomicMinF(float* a, float v) {
    if (v >= 0.f) atomicMin((int*)a, __float_as_int(v));
    else          atomicMax((unsigned int*)a, __float_as_uint(v));
}

// ---------------------------------------------------------------- init mins
__global__ void init_min_kernel(float* __restrict__ p, int n) {
    int i = blockIdx.x * blockDim.x + threadIdx.x;
    if (i < n) p[i] = finf();
}

// ---------------------------------------------------------------- row norms
__global__ __launch_bounds__(256) void norms_kernel(
    const float* __restrict__ x, const float* __restrict__ y,
    float* __restrict__ sqx, float* __restrict__ sqy)
{
    const int wave = threadIdx.x >> 5;
    const int lane = threadIdx.x & 31;
    const int r = blockIdx.x * 8 + wave;            // 0 .. BF_*N_-1
    const float* px = x + (size_t)r * D_;
    const float* py = y + (size_t)r * D_;
    float s1 = 0.f, s2 = 0.f;
    #pragma unroll
    for (int j = lane; j < D_; j += 32) {
        float a = px[j]; s1 += a * a;
        float c = py[j]; s2 += c * c;
    }
    #pragma unroll
    for (int m = 16; m; m >>= 1) {
        s1 += __shfl_xor(s1, m, 32);
        s2 += __shfl_xor(s2, m, 32);
    }
    if (lane == 0) { sqx[r] = s1; sqy[r] = s2; }
}

// ---------------------------------------------------------------- per-(bf,d) min/max over nodes
__global__ __launch_bounds__(256) void minmax_kernel(
    const float* __restrict__ x, const float* __restrict__ y,
    float* __restrict__ mnx, float* __restrict__ mxx,
    float* __restrict__ mny, float* __restrict__ mxy)
{
    const int bf = blockIdx.x;
    const int d  = threadIdx.x;                    // D_ == 256 == blockDim
    const float* px = x + (size_t)bf * N_ * D_ + d;
    const float* py = y + (size_t)bf * N_ * D_ + d;
    float mn1 = finf(), mx1 = -finf(), mn2 = finf(), mx2 = -finf();
    for (int n = 0; n < N_; ++n) {
        float a = px[(size_t)n * D_];
        float c = py[(size_t)n * D_];
        mn1 = fminf(mn1, a); mx1 = fmaxf(mx1, a);
        mn2 = fminf(mn2, c); mx2 = fmaxf(mx2, c);
    }
    const int mm = bf * D_ + d;
    mnx[mm] = mn1; mxx[mm] = mx1;
    mny[mm] = mn2; mxy[mm] = mx2;
}

// ---------------------------------------------------------------- TDM tile load
// 2D tile: 128 rows x 32 f32 out of a 2048 x 256 f32 row-major tensor.
// LDS pad: every 32 DWORDs insert 4 DWORDs -> row stride 36 floats
// (36*l16 mod 64 banks -> conflict-free fragment reads).
__device__ __forceinline__ void tdm_load_tile_128x32(unsigned lds_off,
                                                     const float* gtile)
{
    unsigned long long ga = (unsigned long long)(uintptr_t)gtile;
    v4u g0 = { 1u,                                          // count=1 (valid user D#)
               lds_off,                                     // lds_addr (bytes)
               (unsigned)(ga & 0xFFFFFFFFull),              // global_addr[31:0]
               (unsigned)((ga >> 32) & 0x01FFFFFFull)       // global_addr[56:32]
                   | (2u << 30) };                          // type=2 ("image")
    v8i g1 = { (int)((2u << 16)     // data_size = 4 bytes
                   | (1u << 20)     // pad_enable
                   | (4u << 22)     // pad_interval: 32 DWORDs
                   | (3u << 25)),   // pad_amount : 4 DWORDs
               (int)(256u  << 16),  // tensor_dim0 = 256   (bits 63:48)
               (int)(2048u << 16),  // tensor_dim1 = 2048  (bits 95:80)
               (int)(32u   << 16),  // tile_dim0   = 32    (bits 127:112)
               (int)128,            // tile_dim1   = 128, tile_dim2 = 0
               (int)256,            // tensor_dim0_stride = 256 (low 32 of 48)
               0, 0 };              // stride hi, tensor_dim1_stride = 0
    v4i z4 = { 0, 0, 0, 0 };
#if __has_include(<hip/amd_detail/amd_gfx1250_TDM.h>)
    v8i z8 = { 0, 0, 0, 0, 0, 0, 0, 0 };
    __builtin_amdgcn_tensor_load_to_lds(g0, g1, z4, z4, z8, 0);
#else
    __builtin_amdgcn_tensor_load_to_lds(g0, g1, z4, z4, 0);
#endif
}

// ---------------------------------------------------------------- WMMA gram + row/col mins
// 128x128 tile per block; 8 waves in 2(M)x4(N); each wave 64x32 = 4x2 frags of 16x16.
// K staged through double-buffered LDS by the Tensor Data Mover.
#define TM  128
#define TN  128
#define KC  32
#define LDK 36
#define NCHUNK (D_ / KC)

__global__ __launch_bounds__(256) void gram_min_kernel(
    const float* __restrict__ x, const float* __restrict__ y,
    const float* __restrict__ sqx, const float* __restrict__ sqy,
    float* __restrict__ rmin, float* __restrict__ cmin)
{
    __shared__ float As[2][TM][LDK];
    __shared__ float Bs[2][TN][LDK];

    const int bf  = blockIdx.z;
    const int tm  = blockIdx.y * TM;
    const int tn  = blockIdx.x * TN;
    const int tid = threadIdx.x;
    const int lane = tid & 31;
    const int wave = tid >> 5;
    const int l16  = lane & 15;
    const int lh   = lane >> 4;          // lane-half
    const int wM   = wave & 1;           // 2 waves along M (64 rows each)
    const int wN   = wave >> 1;          // 4 waves along N (32 cols each)

    const float* xt = x + (size_t)bf * N_ * D_ + (size_t)tm * D_;  // A tile row 0
    const float* yt = y + (size_t)bf * N_ * D_ + (size_t)tn * D_;  // B tile row 0
    const unsigned asb = (unsigned)(uintptr_t)(void*)&As[0][0][0];
    const unsigned bsb = (unsigned)(uintptr_t)(void*)&Bs[0][0][0];
    const unsigned bufBytes = TM * LDK * 4;

    // prologue: stage chunk 0 into buffer 0
    if (wave == 0) {
        tdm_load_tile_128x32(asb, xt);
        tdm_load_tile_128x32(bsb, yt);
    }

    v8f acc[4][2];
    {
        v8f z = {};
        #pragma unroll
        for (int mi = 0; mi < 4; ++mi)
            #pragma unroll
            for (int ni = 0; ni < 2; ++ni) acc[mi][ni] = z;
    }

    #pragma unroll 1
    for (int c = 0; c < NCHUNK; ++c) {
        const int buf = c & 1;
        if (wave == 0) {
            if (c + 1 < NCHUNK) {
                // issue next chunk into the other buffer (safe: consumers of it
                // finished before the end-of-iteration barrier of chunk c-1)
                tdm_load_tile_128x32(asb + (unsigned)(buf ^ 1) * bufBytes,
                                     xt + (c + 1) * KC);
                tdm_load_tile_128x32(bsb + (unsigned)(buf ^ 1) * bufBytes,
                                     yt + (c + 1) * KC);
                // TDM completes in order: <=2 outstanding => chunk c resident
                __builtin_amdgcn_s_wait_tensorcnt(2);
            } else {
                __builtin_amdgcn_s_wait_tensorcnt(0);
            }
        }
        __syncthreads();

        #pragma unroll
        for (int ks = 0; ks < KC; ks += 4) {
            const int kk = ks + 2 * lh;  // 16x4 f32 layout: lanes 0-15 K=0,1; 16-31 K=2,3
            v2f afr[4], bfr[2];
            #pragma unroll
            for (int mi = 0; mi < 4; ++mi) {
                const float* p = &As[buf][wM * 64 + mi * 16 + l16][kk];
                v2f t; t.x = p[0]; t.y = p[1];
                afr[mi] = t;
            }
            #pragma unroll
            for (int ni = 0; ni < 2; ++ni) {
                const float* p = &Bs[buf][wN * 32 + ni * 16 + l16][kk];
                v2f t; t.x = p[0]; t.y = p[1];
                bfr[ni] = t;
            }
            #pragma unroll
            for (int mi = 0; mi < 4; ++mi)
                #pragma unroll
                for (int ni = 0; ni < 2; ++ni)
                    acc[mi][ni] = __builtin_amdgcn_wmma_f32_16x16x4_f32(
                        false, afr[mi], false, bfr[ni],
                        (short)0, acc[mi][ni], false, false);
        }
        __syncthreads();   // all waves done with buf before it is re-filled
    }

    // ------- tile reductions: d = a2[m] + b2[n] - 2*acc
    const int n0 = tn + wN * 32;
    const int m0 = tm + wM * 64;

    float b2v[2];
    #pragma unroll
    for (int ni = 0; ni < 2; ++ni) b2v[ni] = sqy[bf * N_ + n0 + ni * 16 + l16];

    float a2v[4][8];
    #pragma unroll
    for (int mi = 0; mi < 4; ++mi)
        #pragma unroll
        for (int v = 0; v < 8; ++v)
            a2v[mi][v] = sqx[bf * N_ + m0 + mi * 16 + lh * 8 + v];

    // column mins (for weight(y,x)): min over m of (a2[m] - 2*acc), + b2[n]
    #pragma unroll
    for (int ni = 0; ni < 2; ++ni) {
        float cm = finf();
        #pragma unroll
        for (int mi = 0; mi < 4; ++mi)
            #pragma unroll
            for (int v = 0; v < 8; ++v)
                cm = fminf(cm, a2v[mi][v] - 2.f * acc[mi][ni][v]);
        cm = fminf(cm, __shfl_xor(cm, 16, 32));   // merge lane-halves (same n)
        if (lh == 0)
            atomicMinF(&cmin[bf * N_ + n0 + ni * 16 + l16], cm + b2v[ni]);
    }

    // row mins (for weight(x,y)): min over n of (b2[n] - 2*acc), + a2[m]
    #pragma unroll
    for (int mi = 0; mi < 4; ++mi) {
        #pragma unroll
        for (int v = 0; v < 8; ++v) {
            float rm = fminf(b2v[0] - 2.f * acc[mi][0][v],
                             b2v[1] - 2.f * acc[mi][1][v]);
            rm = fminf(rm, __shfl_xor(rm, 1, 32));
            rm = fminf(rm, __shfl_xor(rm, 2, 32));
            rm = fminf(rm, __shfl_xor(rm, 4, 32));
            rm = fminf(rm, __shfl_xor(rm, 8, 32));
            const int idx = mi * 8 + v;
            if (l16 == (idx & 15))
                atomicMinF(&rmin[bf * N_ + m0 + mi * 16 + lh * 8 + v],
                           rm + a2v[mi][v]);
        }
    }
}

// ---------------------------------------------------------------- per-batch weights
__global__ __launch_bounds__(256) void wreduce_kernel(
    const float* __restrict__ rmin, const float* __restrict__ cmin,
    float* __restrict__ wx, float* __restrict__ wy)
{
    __shared__ float sx[256], sy[256];
    const int b = blockIdx.x, t = threadIdx.x;
    float s1 = 0.f, s2 = 0.f;
    for (int i = t; i < F_ * N_; i += 256) {
        s1 += rmin[b * F_ * N_ + i];
        s2 += cmin[b * F_ * N_ + i];
    }
    sx[t] = s1; sy[t] = s2;
    __syncthreads();
    for (int o = 128; o > 0; o >>= 1) {
        if (t < o) { sx[t] += sx[t + o]; sy[t] += sy[t + o]; }
        __syncthreads();
    }
    if (t == 0) {
        wx[b] = 1.f / (1.f + sx[0] / (float)(F_ * N_));
        wy[b] = 1.f / (1.f + sy[0] / (float)(F_ * N_));
    }
}

// ---------------------------------------------------------------- final blend
__global__ __launch_bounds__(256) void finalize_kernel(
    const float* __restrict__ x, const float* __restrict__ y,
    const float* __restrict__ mnx, const float* __restrict__ mxx,
    const float* __restrict__ mny, const float* __restrict__ mxy,
    const float* __restrict__ wx, const float* __restrict__ wy,
    float* __restrict__ out)
{
    const size_t i = (size_t)blockIdx.x * 256 + threadIdx.x;
    const int d  = (int)(i & (D_ - 1));
    const int bf = (int)(i / ((size_t)N_ * D_));
    const int b  = bf / F_;
    const int mm = bf * D_ + d;
    const float a = x[i], c = y[i];
    const float nx = (a - mnx[mm]) / (mxx[mm] - mnx[mm]);
    const float ny = (c - mny[mm]) / (mxy[mm] - mny[mm]);
    out[i] = wx[b] * nx + wy[b] * ny;
}

// ---------------------------------------------------------------- launch
extern "C" void kernel_launch(void* const* d_in, const int* in_sizes, int n_in,
                              void* d_out, int out_size, void* d_ws, size_t ws_size,
                              hipStream_t stream) {
    const float* x = (const float*)d_in[0];
    const float* y = (const float*)d_in[1];
    float* out = (float*)d_out;
    float* ws  = (float*)d_ws;

    // workspace layout (floats)
    float* sqx  = ws;                       // BF_*N_
    float* sqy  = sqx  + BF_ * N_;
    float* rmin = sqy  + BF_ * N_;          // BF_*N_
    float* cmin = rmin + BF_ * N_;          // BF_*N_
    float* mnx  = cmin + BF_ * N_;          // BF_*D_
    float* mxx  = mnx  + BF_ * D_;
    float* mny  = mxx  + BF_ * D_;
    float* mxy  = mny  + BF_ * D_;
    float* wx   = mxy  + BF_ * D_;          // B_
    float* wy   = wx   + B_;

    // 1) init row/col mins to +INF (rmin,cmin contiguous)
    {
        int n = 2 * BF_ * N_;
        init_min_kernel<<<(n + 255) / 256, 256, 0, stream>>>(rmin, n);
    }
    // 2) row norms
    norms_kernel<<<BF_ * N_ / 8, 256, 0, stream>>>(x, y, sqx, sqy);
    // 3) per-(bf,d) min/max
    minmax_kernel<<<BF_, 256, 0, stream>>>(x, y, mnx, mxx, mny, mxy);
    // 4) WMMA gram + min reductions (TDM-staged, double-buffered)
    {
        dim3 grid(N_ / TN, N_ / TM, BF_);
        gram_min_kernel<<<grid, 256, 0, stream>>>(x, y, sqx, sqy, rmin, cmin);
    }
    // 5) per-batch weights
    wreduce_kernel<<<B_, 256, 0, stream>>>(rmin, cmin, wx, wy);
    // 6) normalize + blend
    {
        size_t total = (size_t)B_ * F_ * N_ * D_;
        finalize_kernel<<<(unsigned)(total / 256), 256, 0, stream>>>(
            x, y, mnx, mxx, mny, mxy, wx, wy, out);
    }
}